// MoERouter_65687229825641
// MI455X (gfx1250) — compile-verified
//
#include <hip/hip_runtime.h>
#include <hip/hip_bf16.h>
#include <math.h>

typedef __attribute__((ext_vector_type(2))) float v2f;
typedef __attribute__((ext_vector_type(8))) float v8f;

#define D_MODEL 2048
#define N_EXPERTS 8
#define TOP_K 2
#define AUX_LOSS_COEF 0.01f

// ---------------------------------------------------------------------------
// Kernel 0: zero the expert counters in workspace (must run every launch).
// ---------------------------------------------------------------------------
__global__ void moe_init_counts(unsigned int* counts) {
    if (threadIdx.x < N_EXPERTS) counts[threadIdx.x] = 0u;
}

// ---------------------------------------------------------------------------
// Kernel 1: gate GEMM via V_WMMA_F32_16X16X4_F32 + softmax + top-2 routing.
// Block = 256 threads = 8 waves; each wave computes logits for a 16-token
// tile (M=16 tokens, N=16 lanes with experts 0..7 valid, K=2048 in steps of 4).
// ---------------------------------------------------------------------------
__global__ void __launch_bounds__(256)
moe_router_main(const float* __restrict__ hidden,   // [N_tokens, D_MODEL]
                const float* __restrict__ W,        // [N_EXPERTS, D_MODEL]
                float* __restrict__ out,            // concat output
                unsigned int* __restrict__ counts,  // [N_EXPERTS]
                int n_tokens) {
    __shared__ float lds_logits[128][N_EXPERTS];    // 4 KB per block

    const int lane = threadIdx.x & 31;
    const int wave = threadIdx.x >> 5;

    // ---- Phase 1: WMMA gate GEMM -------------------------------------------
    const int tile = blockIdx.x * 8 + wave;         // 16-token tile index
    const int m    = lane & 15;                     // token row within tile / expert col
    const int kdel = (lane >> 4) << 1;              // lanes 16-31 hold K+2,K+3

    const float* hp = hidden + (size_t)(tile * 16 + m) * D_MODEL + kdel;
    // B fragment: N = lane&15 is the expert index; experts 8..15 are zero-padded.
    const float  bmask = (m < N_EXPERTS) ? 1.0f : 0.0f;
    const float* wp = W + (size_t)(m & (N_EXPERTS - 1)) * D_MODEL + kdel;

    v8f c = {};
#pragma unroll 8
    for (int k0 = 0; k0 < D_MODEL; k0 += 4) {
        v2f a = *(const v2f*)(hp + k0);             // global_load_b64, coalesced
        v2f b = *(const v2f*)(wp + k0);             // W stays hot in L2/WGP$
        b.x *= bmask;
        b.y *= bmask;
        // D = A(16x4) * B(4x16) + C  : v_wmma_f32_16x16x4_f32
        c = __builtin_amdgcn_wmma_f32_16x16x4_f32(
            /*neg_a=*/false, a, /*neg_b=*/false, b,
            /*c_mod=*/(short)0, c, /*reuse_a=*/false, /*reuse_b=*/false);
    }

    // Scatter logits to LDS. Accumulator layout: VGPR j holds row M=j (lanes
    // 0-15) or M=j+8 (lanes 16-31); N (expert) = lane&15.
    if (m < N_EXPERTS) {
        const int trow = wave * 16 + ((lane >> 4) << 3);
#pragma unroll
        for (int j = 0; j < 8; ++j)
            lds_logits[trow + j][m] = c[j];
    }
    __syncthreads();

    // ---- Phase 2: softmax + top-2 per token --------------------------------
    if (threadIdx.x < 128) {
        const int t     = threadIdx.x;
        const int token = blockIdx.x * 128 + t;

        float l[N_EXPERTS];
#pragma unroll
        for (int e = 0; e < N_EXPERTS; ++e) l[e] = lds_logits[t][e];

        float mx = l[0];
#pragma unroll
        for (int e = 1; e < N_EXPERTS; ++e) mx = fmaxf(mx, l[e]);

        float p[N_EXPERTS];
        float s = 0.0f;
#pragma unroll
        for (int e = 0; e < N_EXPERTS; ++e) { p[e] = __expf(l[e] - mx); s += p[e]; }
        const float inv = 1.0f / s;
#pragma unroll
        for (int e = 0; e < N_EXPERTS; ++e) p[e] *= inv;

        // top-2, first occurrence wins on ties (matches jax.lax.top_k)
        int   i0 = 0;
        float v0 = p[0];
#pragma unroll
        for (int e = 1; e < N_EXPERTS; ++e)
            if (p[e] > v0) { v0 = p[e]; i0 = e; }
        int   i1 = -1;
        float v1 = -INFINITY;
#pragma unroll
        for (int e = 0; e < N_EXPERTS; ++e)
            if (e != i0 && p[e] > v1) { v1 = p[e]; i1 = e; }

        // outputs: indices [N,2] | probs [N,2] | aux [1] | load [8]
        out[token * 2 + 0] = (float)i0;
        out[token * 2 + 1] = (float)i1;
        float* probs = out + (size_t)2 * n_tokens;
        probs[token * 2 + 0] = v0;
        probs[token * 2 + 1] = v1;

        atomicAdd(&counts[i0], 1u);
        atomicAdd(&counts[i1], 1u);
    }
}

// ---------------------------------------------------------------------------
// Kernel 2: load distribution + aux loss from the integer counters.
// ---------------------------------------------------------------------------
__global__ void moe_router_finalize(const unsigned int* __restrict__ counts,
                                    float* __restrict__ out, int n_tokens) {
    if (threadIdx.x == 0 && blockIdx.x == 0) {
        const float denom = (float)(n_tokens * TOP_K);
        float load[N_EXPERTS];
        float sum = 0.0f;
#pragma unroll
        for (int e = 0; e < N_EXPERTS; ++e) {
            load[e] = (float)counts[e] / denom;
            sum += load[e];
        }
        const float mean = sum / (float)N_EXPERTS;
        float var = 0.0f;
#pragma unroll
        for (int e = 0; e < N_EXPERTS; ++e) {
            const float d = load[e] - mean;
            var += d * d;
        }
        var /= (float)N_EXPERTS;
        const float cv_sq = var / (mean * mean + 1e-9f);

        float* tail = out + (size_t)4 * n_tokens;
        tail[0] = AUX_LOSS_COEF * cv_sq;
#pragma unroll
        for (int e = 0; e < N_EXPERTS; ++e) tail[1 + e] = load[e];
    }
}

// ---------------------------------------------------------------------------
extern "C" void kernel_launch(void* const* d_in, const int* in_sizes, int n_in,
                              void* d_out, int out_size, void* d_ws, size_t ws_size,
                              hipStream_t stream) {
    const float* hidden = (const float*)d_in[0];   // [4,4096,2048] f32
    const float* W      = (const float*)d_in[1];   // [8,2048] f32
    float* out          = (float*)d_out;
    unsigned int* counts = (unsigned int*)d_ws;

    const int n_tokens = in_sizes[0] / D_MODEL;    // 16384
    const int blocks   = n_tokens / 128;           // 128 tokens per block

    moe_init_counts<<<1, 32, 0, stream>>>(counts);
    moe_router_main<<<blocks, 256, 0, stream>>>(hidden, W, out, counts, n_tokens);
    moe_router_finalize<<<1, 32, 0, stream>>>(counts, out, n_tokens);
}